// MessagePassingMultiQuant_20418274525751
// MI455X (gfx1250) — compile-verified
//
#include <hip/hip_runtime.h>
#include <hip/hip_bf16.h>
#include <stdint.h>

#define N_NODES 100000
#define N_EDGES 1000000
#define D_FEAT  64
#define QMAXF   127.0f
#define E_TILE  256

// fake-quant one value: clip(round(v/s), -128, 127) * s
__device__ __forceinline__ float dq(float v, float s) {
    float r = rintf(v / s);                 // round-to-nearest-even == jnp.round
    r = fmaxf(r, -(QMAXF + 1.0f));
    r = fminf(r, QMAXF);
    return r * s;
}

__device__ __forceinline__ float absmax4(float4 v) {
    return fmaxf(fmaxf(fabsf(v.x), fabsf(v.y)), fmaxf(fabsf(v.z), fabsf(v.w)));
}

__device__ __forceinline__ float scale_from_bits(unsigned int bits) {
    return fmaxf(__uint_as_float(bits) / QMAXF, 1e-8f);
}

// CDNA5 async copy: global (4B per lane) -> LDS, tracked by ASYNCcnt.
// GV mode: %0 = VGPR LDS byte address, %1 = VGPR pair 64-bit global address.
// LDS byte address = low 32 bits of the generic pointer (aperture is in [63:32]).
__device__ __forceinline__ void async_g2l_b32(const int* gptr, int* lptr) {
    uint32_t lds_addr = (uint32_t)(uintptr_t)lptr;
    uint64_t gaddr    = (uint64_t)(uintptr_t)gptr;
    asm volatile("global_load_async_to_lds_b32 %0, %1, off"
                 :: "v"(lds_addr), "v"(gaddr) : "memory");
}

__device__ __forceinline__ void wait_asynccnt0() {
    asm volatile("s_wait_asynccnt 0x0" ::: "memory");
}

// K0: zero output accumulator + amax scalars (runs every call; harness poisons once)
__global__ void k_init(float4* __restrict__ out4, unsigned int* __restrict__ amax) {
    int i = blockIdx.x * blockDim.x + threadIdx.x;
    if (i < (N_NODES * D_FEAT) / 4) out4[i] = make_float4(0.f, 0.f, 0.f, 0.f);
    if (blockIdx.x == 0 && threadIdx.x < 4) amax[threadIdx.x] = 0u;
}

// K1: per-node row abs-max. 16 lanes per node, float4 per lane (256B/row coalesced).
__global__ void k_rowmax(const float* __restrict__ x, float* __restrict__ rowmax) {
    int node = blockIdx.x * 16 + (threadIdx.x >> 4);
    int l    = threadIdx.x & 15;
    float m  = 0.f;
    if (node < N_NODES)
        m = absmax4(((const float4*)x)[node * 16 + l]);
    // half-wave (16-lane) reduce; wave32 on gfx1250 keeps xor<16 inside the group
    for (int off = 8; off; off >>= 1)
        m = fmaxf(m, __shfl_xor(m, off, 32));
    if (l == 0 && node < N_NODES) rowmax[node] = m;
}

// K2: amax over unprotected gathered edge rows == max(rowmax[src] | !mask[src])
__global__ void k_edge_amax(const int* __restrict__ src, const int* __restrict__ mask,
                            const float* __restrict__ rowmax, unsigned int* __restrict__ amax) {
    int tid    = blockIdx.x * blockDim.x + threadIdx.x;
    int stride = gridDim.x * blockDim.x;
    float lm = 0.f;
    for (int e = tid; e < N_EDGES; e += stride) {
        __builtin_prefetch(src + e + 4096, 0, 0);   // global_prefetch_b8
        int s = src[e];
        if (!mask[s]) lm = fmaxf(lm, rowmax[s]);
    }
    for (int off = 16; off; off >>= 1)
        lm = fmaxf(lm, __shfl_xor(lm, off, 32));
    if ((threadIdx.x & 31) == 0)
        atomicMax(amax + 0, __float_as_uint(lm));   // values >= 0: uint order == float order
}

// K3: gather x[src], quantize unprotected edges, scatter-add to out[dst].
// Each block stages a 256-edge tile of src/dst into LDS via async-to-LDS copies
// (ASYNCcnt path), then 16-lane groups process edges with float4 gathers and
// 4x global_atomic_add_f32 per lane.
__global__ void k_scatter(const float* __restrict__ x, const int* __restrict__ src,
                          const int* __restrict__ dst, const int* __restrict__ mask,
                          const unsigned int* __restrict__ amax, float* __restrict__ out) {
    __shared__ int s_src[E_TILE];
    __shared__ int s_dst[E_TILE];

    float s1  = scale_from_bits(amax[0]);
    int t     = threadIdx.x;
    int l     = t & 15;          // lane within edge-group (float4 column)
    int grp   = t >> 4;          // 16 edge-groups per 256-thread block
    int base  = blockIdx.x * E_TILE;
    int limit = N_EDGES - base;
    if (limit > E_TILE) limit = E_TILE;

    // ---- async stage: one b32 per lane for src and dst (EXEC-masked tail) ----
    if (t < limit) {
        async_g2l_b32(src + base + t, s_src + t);   // global_load_async_to_lds_b32
        async_g2l_b32(dst + base + t, s_dst + t);
    }
    wait_asynccnt0();            // wave-local ASYNCcnt drain ...
    __syncthreads();             // ... then workgroup barrier -> all LDS writes visible

    for (int k = grp; k < limit; k += 16) {
        int s    = s_src[k];     // ds_load broadcast across the 16-lane group
        int d    = s_dst[k];
        int prot = mask[s];
        float4 v = ((const float4*)x)[s * 16 + l];
        if (!prot) {
            v.x = dq(v.x, s1); v.y = dq(v.y, s1);
            v.z = dq(v.z, s1); v.w = dq(v.w, s1);
        }
        float* o = out + d * D_FEAT + l * 4;
        atomicAdd(o + 0, v.x);   // global_atomic_add_f32, L2-resident accumulator
        atomicAdd(o + 1, v.y);
        atomicAdd(o + 2, v.z);
        atomicAdd(o + 3, v.w);
    }
}

// K4: amax over unprotected rows of aggr (LDS block reduce -> 1 atomic/block)
__global__ void k_aggr_amax(const float* __restrict__ out, const int* __restrict__ mask,
                            unsigned int* __restrict__ amax) {
    __shared__ float red[16];
    int node = blockIdx.x * 16 + (threadIdx.x >> 4);
    int l    = threadIdx.x & 15;
    float m  = 0.f;
    if (node < N_NODES && !mask[node])
        m = absmax4(((const float4*)out)[node * 16 + l]);
    for (int off = 8; off; off >>= 1)
        m = fmaxf(m, __shfl_xor(m, off, 32));
    if (l == 0) red[threadIdx.x >> 4] = m;
    __syncthreads();
    if (threadIdx.x == 0) {
        float bm = red[0];
        #pragma unroll
        for (int i = 1; i < 16; ++i) bm = fmaxf(bm, red[i]);
        atomicMax(amax + 1, __float_as_uint(bm));
    }
}

// K5: quantize unprotected rows in place (scale2), fused amax of the quantized result
// (needed separately: clip lower bound -128 can change the abs-max)
__global__ void k_quant_a(float* __restrict__ out, const int* __restrict__ mask,
                          unsigned int* __restrict__ amax) {
    __shared__ float red[16];
    float s2 = scale_from_bits(amax[1]);
    int node = blockIdx.x * 16 + (threadIdx.x >> 4);
    int l    = threadIdx.x & 15;
    float m  = 0.f;
    if (node < N_NODES && !mask[node]) {
        float4 v = ((const float4*)out)[node * 16 + l];
        v.x = dq(v.x, s2); v.y = dq(v.y, s2);
        v.z = dq(v.z, s2); v.w = dq(v.w, s2);
        ((float4*)out)[node * 16 + l] = v;
        m = absmax4(v);
    }
    for (int off = 8; off; off >>= 1)
        m = fmaxf(m, __shfl_xor(m, off, 32));
    if (l == 0) red[threadIdx.x >> 4] = m;
    __syncthreads();
    if (threadIdx.x == 0) {
        float bm = red[0];
        #pragma unroll
        for (int i = 1; i < 16; ++i) bm = fmaxf(bm, red[i]);
        atomicMax(amax + 2, __float_as_uint(bm));
    }
}

// K6: final quant pass (scale3) in place
__global__ void k_quant_b(float* __restrict__ out, const int* __restrict__ mask,
                          const unsigned int* __restrict__ amax) {
    float s3 = scale_from_bits(amax[2]);
    int node = blockIdx.x * 16 + (threadIdx.x >> 4);
    int l    = threadIdx.x & 15;
    if (node < N_NODES && !mask[node]) {
        float4 v = ((const float4*)out)[node * 16 + l];
        v.x = dq(v.x, s3); v.y = dq(v.y, s3);
        v.z = dq(v.z, s3); v.w = dq(v.w, s3);
        ((float4*)out)[node * 16 + l] = v;
    }
}

extern "C" void kernel_launch(void* const* d_in, const int* in_sizes, int n_in,
                              void* d_out, int out_size, void* d_ws, size_t ws_size,
                              hipStream_t stream) {
    const float* x    = (const float*)d_in[0];
    const int*   ei   = (const int*)d_in[1];   // [2, N_EDGES] row-major
    const int*   mask = (const int*)d_in[2];   // bool as int32
    const int*   src  = ei;
    const int*   dst  = ei + N_EDGES;
    float*       out  = (float*)d_out;

    // ws layout: [0,256) -> 4 amax uint slots; [256, 256+4*N_NODES) -> rowmax
    unsigned int* amax   = (unsigned int*)d_ws;
    float*        rowmax = (float*)((char*)d_ws + 256);

    const int nodeBlocks = (N_NODES + 15) / 16;          // 16 nodes per 256-thread block
    const int edgeBlocks = (N_EDGES + E_TILE - 1) / E_TILE;

    k_init<<<((N_NODES * D_FEAT) / 4 + 255) / 256, 256, 0, stream>>>((float4*)out, amax);
    k_rowmax<<<nodeBlocks, 256, 0, stream>>>(x, rowmax);
    k_edge_amax<<<1024, 256, 0, stream>>>(src, mask, rowmax, amax);
    k_scatter<<<edgeBlocks, 256, 0, stream>>>(x, src, dst, mask, amax, out);
    k_aggr_amax<<<nodeBlocks, 256, 0, stream>>>(out, mask, amax);
    k_quant_a<<<nodeBlocks, 256, 0, stream>>>(out, mask, amax);
    k_quant_b<<<nodeBlocks, 256, 0, stream>>>(out, mask, amax);
}